// GraphEncoder_42004780154958
// MI455X (gfx1250) — compile-verified
//
#include <hip/hip_runtime.h>
#include <hip/hip_bf16.h>

typedef __attribute__((ext_vector_type(2))) float v2f;
typedef __attribute__((ext_vector_type(8))) float v8f;

namespace {
constexpr int F_INC  = 128;
constexpr int HCH    = 256;   // graph hidden channels
constexpr int NHIDC  = 512;
constexpr int NOUTC  = 256;
constexpr int GNUM   = 256;   // graphs in batch
}

// ---------------------------------------------------------------------------
// Degree / dinv
// ---------------------------------------------------------------------------
__global__ void deg_accum_kernel(const int* __restrict__ dst, float* __restrict__ deg, int E) {
    int e = blockIdx.x * blockDim.x + threadIdx.x;
    if (e < E) atomicAdd(&deg[dst[e]], 1.0f);
}

__global__ void dinv_kernel(float* __restrict__ deg, int n) {
    int i = blockIdx.x * blockDim.x + threadIdx.x;
    if (i < n) deg[i] = rsqrtf(1.0f + deg[i]);   // deg^{-1/2}, self-loop included
}

// ---------------------------------------------------------------------------
// f32 WMMA GEMM: C[M,Nc] = A[M,K] @ B[K,Nc] (+bias)(+relu)
// Block = 512 threads = 16 waves; each wave owns one 16x16 C tile.
// The 16xK A row-tile is staged once into LDS (shared by all 16 waves),
// then WMMA A-fragments come from LDS (ds_load_b64, bank-conflict-free via
// +4 float row padding). B (weights) stays in global: <=512KB, L2-resident.
// grid.x = M/16, grid.y = Nc/256. Requires M%16==0, K%4==0, Nc%256==0.
// ---------------------------------------------------------------------------
template <int K>
__global__ __launch_bounds__(512)
void gemm_wmma_f32(const float* __restrict__ A, const float* __restrict__ B,
                   float* __restrict__ C, int Nc,
                   const float* __restrict__ bias, int do_relu) {
    constexpr int LDK = K + 4;              // pad: rows start on bank 4m (mod 64)
    __shared__ float As[16 * LDK];

    const int tid  = threadIdx.x;
    const int lane = tid & 31;
    const int wave = tid >> 5;              // 0..15
    const int row0 = blockIdx.x * 16;

    // Cooperative stage of the 16xK A tile into LDS (float4 chunks, exact fit:
    // 16*K/4 is a multiple of 512 for K in {128,256,512} -> no divergence).
    constexpr int CH = 16 * K / 4;
#pragma unroll
    for (int idx = tid; idx < CH; idx += 512) {
        int r  = idx / (K / 4);
        int c4 = (idx % (K / 4)) * 4;
        float4 v = *(const float4*)(A + (long)(row0 + r) * K + c4);
        *(float4*)(&As[r * LDK + c4]) = v;
    }
    __syncthreads();

    const int col0 = (blockIdx.y * 16 + wave) * 16;
    const int m    = lane & 15;             // row (A) / col (B,C) within tile
    const int koff = (lane >> 4) * 2;       // lanes 16-31 handle K+2,K+3
    const int aoff = m * LDK + koff;

    v8f acc = {};
    const float* __restrict__ bcol = B + (long)koff * Nc + (col0 + m);

    for (int k = 0; k < K; k += 4) {
        // A fragment from LDS: lane<16 -> As[m][k..k+1], lane>=16 -> As[m][k+2..k+3]
        v2f a = *(const v2f*)(&As[aoff + k]);
        // B fragment: lane<16 -> B[k..k+1][col0+m], lane>=16 -> B[k+2..k+3][col0+m]
        v2f b;
        b.x = bcol[(long)k * Nc];
        b.y = bcol[(long)(k + 1) * Nc];
        acc = __builtin_amdgcn_wmma_f32_16x16x4_f32(
            /*neg_a=*/false, a, /*neg_b=*/false, b,
            /*c_mod=*/(short)0, acc, /*reuse_a=*/false, /*reuse_b=*/false);
    }

    // C/D layout: VGPR v, lanes 0-15 -> (M=v, N=lane); lanes 16-31 -> (M=v+8, N=lane-16)
    const int rbase = row0 + ((lane >> 4) ? 8 : 0);
    const int c     = col0 + m;
    const float bv  = bias ? bias[c] : 0.0f;
#pragma unroll
    for (int v = 0; v < 8; ++v) {
        float val = acc[v] + bv;
        if (do_relu) val = fmaxf(val, 0.0f);
        C[(long)(rbase + v) * Nc + c] = val;
    }
}

// ---------------------------------------------------------------------------
// Edge scatter: agg[dst] += h[src] * dinv[src]*dinv[dst]
// 64 threads per edge, float4 per thread (H=256). 4 edges per 256-thread block.
// ---------------------------------------------------------------------------
__global__ __launch_bounds__(256)
void edge_scatter_kernel(const float* __restrict__ h, const int* __restrict__ src,
                         const int* __restrict__ dst, const float* __restrict__ dinv,
                         float* __restrict__ agg, int E) {
    int e = blockIdx.x * 4 + (threadIdx.x >> 6);
    if (e >= E) return;
    int f = (threadIdx.x & 63) * 4;
    int s = src[e];
    int d = dst[e];
    float norm = dinv[s] * dinv[d];
    const float4 hv = *(const float4*)(h + (long)s * HCH + f);
    float* out = agg + (long)d * HCH + f;
    atomicAdd(out + 0, hv.x * norm);
    atomicAdd(out + 1, hv.y * norm);
    atomicAdd(out + 2, hv.z * norm);
    atomicAdd(out + 3, hv.w * norm);
}

// ---------------------------------------------------------------------------
// Finalize layer: out = agg + h*dinv^2 + b  (optional relu). float4 vectorized.
// out may alias agg or h (same-index elementwise).
// ---------------------------------------------------------------------------
__global__ void finalize_kernel(const float* __restrict__ agg, const float* __restrict__ h,
                                const float* __restrict__ dinv, const float* __restrict__ bias,
                                float* __restrict__ out, int n4, int do_relu) {
    int i = blockIdx.x * blockDim.x + threadIdx.x;   // float4 index
    if (i >= n4) return;
    int node = (i * 4) / HCH;
    int f    = (i * 4) % HCH;
    float di = dinv[node];
    float s  = di * di;
    float4 a  = ((const float4*)agg)[i];
    float4 hv = ((const float4*)h)[i];
    float4 r;
    r.x = a.x + hv.x * s + bias[f + 0];
    r.y = a.y + hv.y * s + bias[f + 1];
    r.z = a.z + hv.z * s + bias[f + 2];
    r.w = a.w + hv.w * s + bias[f + 3];
    if (do_relu) {
        r.x = fmaxf(r.x, 0.0f); r.y = fmaxf(r.y, 0.0f);
        r.z = fmaxf(r.z, 0.0f); r.w = fmaxf(r.w, 0.0f);
    }
    ((float4*)out)[i] = r;
}

// ---------------------------------------------------------------------------
// Mean pooling
// ---------------------------------------------------------------------------
__global__ void count_kernel(const int* __restrict__ batch, float* __restrict__ counts, int n) {
    int i = blockIdx.x * blockDim.x + threadIdx.x;
    if (i < n) atomicAdd(&counts[batch[i]], 1.0f);
}

__global__ void pool_accum_kernel(const float* __restrict__ h, const int* __restrict__ batch,
                                  float* __restrict__ pooled, int n4) {
    int i = blockIdx.x * blockDim.x + threadIdx.x;   // float4 index
    if (i >= n4) return;
    int node = (i * 4) / HCH;
    int f    = (i * 4) % HCH;
    int g = batch[node];
    float4 hv = ((const float4*)h)[i];
    float* o = pooled + (long)g * HCH + f;
    atomicAdd(o + 0, hv.x);
    atomicAdd(o + 1, hv.y);
    atomicAdd(o + 2, hv.z);
    atomicAdd(o + 3, hv.w);
}

__global__ void pool_div_kernel(float* __restrict__ pooled, const float* __restrict__ counts, int n) {
    int i = blockIdx.x * blockDim.x + threadIdx.x;
    if (i < n) pooled[i] /= fmaxf(counts[i / HCH], 1.0f);
}

// ---------------------------------------------------------------------------
// Host side
// ---------------------------------------------------------------------------
extern "C" void kernel_launch(void* const* d_in, const int* in_sizes, int n_in,
                              void* d_out, int out_size, void* d_ws, size_t ws_size,
                              hipStream_t stream) {
    const float* x     = (const float*)d_in[0];
    const int*   src   = (const int*)  d_in[1];
    const int*   dst   = (const int*)  d_in[2];
    const int*   batch = (const int*)  d_in[3];
    const float* W1    = (const float*)d_in[4];
    const float* b1    = (const float*)d_in[5];
    const float* W2    = (const float*)d_in[6];
    const float* b2    = (const float*)d_in[7];
    const float* W3    = (const float*)d_in[8];
    const float* b3    = (const float*)d_in[9];
    const float* Wm1   = (const float*)d_in[10];
    const float* bm1   = (const float*)d_in[11];
    const float* Wm2   = (const float*)d_in[12];
    const float* bm2   = (const float*)d_in[13];

    const int N = in_sizes[0] / F_INC;   // 100000
    const int E = in_sizes[1];           // 1600000
    float* out = (float*)d_out;          // [G, NOUT] f32

    // Workspace carve-up (16B aligned chunks)
    char* ws = (char*)d_ws;
    auto align16 = [](size_t v) { return (v + 15) & ~(size_t)15; };
    float* dinv   = (float*)ws;                    ws += align16((size_t)N * sizeof(float));
    float* bufA   = (float*)ws;                    ws += align16((size_t)N * HCH * sizeof(float));
    float* bufB   = (float*)ws;                    ws += align16((size_t)N * HCH * sizeof(float));
    float* pooled = (float*)ws;                    ws += align16((size_t)GNUM * HCH * sizeof(float));
    float* counts = (float*)ws;                    ws += align16((size_t)GNUM * sizeof(float));
    float* t1     = (float*)ws;                    ws += align16((size_t)GNUM * NHIDC * sizeof(float));
    (void)ws_size; (void)n_in; (void)out_size;

    const int TB = 256;
    const size_t nodeBytes = (size_t)N * HCH * sizeof(float);
    const int n4 = N * HCH / 4;

    // --- degrees -> dinv (in place) ---
    hipMemsetAsync(dinv, 0, (size_t)N * sizeof(float), stream);
    deg_accum_kernel<<<(E + TB - 1) / TB, TB, 0, stream>>>(dst, dinv, E);
    dinv_kernel<<<(N + TB - 1) / TB, TB, 0, stream>>>(dinv, N);

    dim3 blk(TB);
    dim3 gblk(512);                      // GEMM blocks: 16 waves
    dim3 gN16(N / 16, HCH / 256);        // node GEMMs, Nc=256 -> grid.y=1
    dim3 gM1(GNUM / 16, NHIDC / 256);    // MLP1, Nc=512 -> grid.y=2
    dim3 gM2(GNUM / 16, NOUTC / 256);    // MLP2, Nc=256 -> grid.y=1

    // --- Layer 1: h = x@W1 ; agg ; relu(agg + h*dinv^2 + b1) -> bufA ---
    gemm_wmma_f32<F_INC><<<gN16, gblk, 0, stream>>>(x, W1, bufA, HCH, nullptr, 0);
    hipMemsetAsync(bufB, 0, nodeBytes, stream);
    edge_scatter_kernel<<<(E + 3) / 4, blk, 0, stream>>>(bufA, src, dst, dinv, bufB, E);
    finalize_kernel<<<(n4 + TB - 1) / TB, blk, 0, stream>>>(bufB, bufA, dinv, b1, bufA, n4, 1);

    // --- Layer 2 ---
    gemm_wmma_f32<HCH><<<gN16, gblk, 0, stream>>>(bufA, W2, bufB, HCH, nullptr, 0);
    hipMemsetAsync(bufA, 0, nodeBytes, stream);
    edge_scatter_kernel<<<(E + 3) / 4, blk, 0, stream>>>(bufB, src, dst, dinv, bufA, E);
    finalize_kernel<<<(n4 + TB - 1) / TB, blk, 0, stream>>>(bufA, bufB, dinv, b2, bufB, n4, 1);

    // --- Layer 3 (no relu) ---
    gemm_wmma_f32<HCH><<<gN16, gblk, 0, stream>>>(bufB, W3, bufA, HCH, nullptr, 0);
    hipMemsetAsync(bufB, 0, nodeBytes, stream);
    edge_scatter_kernel<<<(E + 3) / 4, blk, 0, stream>>>(bufA, src, dst, dinv, bufB, E);
    finalize_kernel<<<(n4 + TB - 1) / TB, blk, 0, stream>>>(bufB, bufA, dinv, b3, bufA, n4, 0);
    // bufA = h3

    // --- mean pool ---
    hipMemsetAsync(pooled, 0, (size_t)GNUM * HCH * sizeof(float), stream);
    hipMemsetAsync(counts, 0, (size_t)GNUM * sizeof(float), stream);
    count_kernel<<<(N + TB - 1) / TB, blk, 0, stream>>>(batch, counts, N);
    pool_accum_kernel<<<(n4 + TB - 1) / TB, blk, 0, stream>>>(bufA, batch, pooled, n4);
    pool_div_kernel<<<(GNUM * HCH + TB - 1) / TB, blk, 0, stream>>>(pooled, counts, GNUM * HCH);

    // --- MLP: relu(pooled@Wm1+bm1) @ Wm2 + bm2 ---
    gemm_wmma_f32<HCH><<<gM1, gblk, 0, stream>>>(pooled, Wm1, t1, NHIDC, bm1, 1);
    gemm_wmma_f32<NHIDC><<<gM2, gblk, 0, stream>>>(t1, Wm2, out, NOUTC, bm2, 0);
}